// SEGNN_61692910240303
// MI455X (gfx1250) — compile-verified
//
#include <hip/hip_runtime.h>
#include <math.h>

// ---------------------------------------------------------------------------
// SEGNN message passing on MI455X (gfx1250, wave32).
// Edge MLP GEMMs run on V_WMMA_F32_16X16X4_F32 (fp32 matrix pipe).
// ---------------------------------------------------------------------------

typedef float v2f __attribute__((ext_vector_type(2)));
typedef float v8f __attribute__((ext_vector_type(8)));

#define Hh 16            // hidden multiplicity
#define WAVES 4          // waves per edge block (16 edges per wave)
#define SPAD 33          // padded row stride for LDS scratch (odd -> conflict-free A reads)

__device__ inline v8f wmma4(v2f a, v2f b, v8f c) {
  // D = A(16x4, f32) * B(4x16, f32) + C(16x16, f32)
  return __builtin_amdgcn_wmma_f32_16x16x4_f32(
      /*neg_a=*/false, a, /*neg_b=*/false, b,
      /*c_mod=*/(short)0, c, /*reuse_a=*/false, /*reuse_b=*/false);
}

__device__ inline float sigmoid_f(float x) { return 1.0f / (1.0f + __expf(-x)); }

__device__ inline void copy_lds(float* dst, const float* __restrict__ src,
                                int n, int tid, int bdim) {
  for (int i = tid; i < n; i += bdim) dst[i] = src[i];
}

// ---------------------------------------------------------------------------
// One TP+Linear(+Gate) edge layer for a 16-edge tile, one wave.
//   s_in : [16][SPAD]      ms features (row = edge, col = feature)
//   v_in : [3][16][SPAD]   mv features per cartesian component d
//   Ws   : [2*FIN][FS], Wv : [2*FIN][16] in LDS; bsv: [FS]
// Decomposition:
//   s_out      = es*(ms@Ws_top) + sum_d ev_d*(mv_d@Ws_bot) + b
//   v_out_d    = ev_d*(ms@Wv_top) + es*(mv_d@Wv_bot)
// GATED: gates = sigmoid(s_out[:, :16]); scalars = silu(s_out[:, 16:32]);
//        vectors *= gates.
// ---------------------------------------------------------------------------
template <int FIN, int FS, bool GATED, bool WRITEBACK>
__device__ void edge_layer(const float* __restrict__ Ws,
                           const float* __restrict__ bsv,
                           const float* __restrict__ Wv,
                           float* s_in, float* v_in,
                           const float* es, const float* ev,
                           int lane, v8f& outS, v8f (&outV)[3]) {
  static_assert(FIN % 4 == 0, "K must be multiple of 4");
  const int row   = lane & 15;         // A row (edge) / B,D column (channel)
  const int khalf = (lane >> 4) * 2;   // which K pair this half-wave holds
  const int mbase = (lane >> 4) * 8;   // D-layout M base for this half-wave

  // per-M (per-edge) steering scalars for D-accumulator combine
  float esM[8], evM[8][3];
#pragma unroll
  for (int j = 0; j < 8; ++j) {
    const int M = mbase + j;
    esM[j] = es[M];
    evM[j][0] = ev[M * 3 + 0];
    evM[j][1] = ev[M * 3 + 1];
    evM[j][2] = ev[M * 3 + 2];
  }

  constexpr int NT = FS / 16;
  v8f sOut[NT];
#pragma unroll
  for (int nt = 0; nt < NT; ++nt) {
    // P = ms @ Ws_top
    v8f acc = {0.f, 0.f, 0.f, 0.f, 0.f, 0.f, 0.f, 0.f};
#pragma unroll
    for (int c = 0; c < FIN / 4; ++c) {
      const int k = 4 * c + khalf;
      v2f a; a.x = s_in[row * SPAD + k]; a.y = s_in[row * SPAD + k + 1];
      v2f b; b.x = Ws[k * FS + nt * 16 + row]; b.y = Ws[(k + 1) * FS + nt * 16 + row];
      acc = wmma4(a, b, acc);
    }
    v8f res;
#pragma unroll
    for (int j = 0; j < 8; ++j) res[j] = esM[j] * acc[j];
    // + sum_d ev_d * (mv_d @ Ws_bot)
#pragma unroll
    for (int d = 0; d < 3; ++d) {
      v8f q = {0.f, 0.f, 0.f, 0.f, 0.f, 0.f, 0.f, 0.f};
#pragma unroll
      for (int c = 0; c < FIN / 4; ++c) {
        const int k = 4 * c + khalf;
        v2f a; a.x = v_in[(d * 16 + row) * SPAD + k];
               a.y = v_in[(d * 16 + row) * SPAD + k + 1];
        v2f b; b.x = Ws[(FIN + k) * FS + nt * 16 + row];
               b.y = Ws[(FIN + k + 1) * FS + nt * 16 + row];
        q = wmma4(a, b, q);
      }
#pragma unroll
      for (int j = 0; j < 8; ++j) res[j] += evM[j][d] * q[j];
    }
    const float bias = bsv[nt * 16 + row];
#pragma unroll
    for (int j = 0; j < 8; ++j) res[j] += bias;
    sOut[nt] = res;
  }

  // vector outputs (always 16 channels): R = ms@Wv_top ; S_d = mv_d@Wv_bot
  v8f vTmp[3];
  {
    v8f r = {0.f, 0.f, 0.f, 0.f, 0.f, 0.f, 0.f, 0.f};
#pragma unroll
    for (int c = 0; c < FIN / 4; ++c) {
      const int k = 4 * c + khalf;
      v2f a; a.x = s_in[row * SPAD + k]; a.y = s_in[row * SPAD + k + 1];
      v2f b; b.x = Wv[k * 16 + row]; b.y = Wv[(k + 1) * 16 + row];
      r = wmma4(a, b, r);
    }
#pragma unroll
    for (int d = 0; d < 3; ++d) {
      v8f s2 = {0.f, 0.f, 0.f, 0.f, 0.f, 0.f, 0.f, 0.f};
#pragma unroll
      for (int c = 0; c < FIN / 4; ++c) {
        const int k = 4 * c + khalf;
        v2f a; a.x = v_in[(d * 16 + row) * SPAD + k];
               a.y = v_in[(d * 16 + row) * SPAD + k + 1];
        v2f b; b.x = Wv[(FIN + k) * 16 + row]; b.y = Wv[(FIN + k + 1) * 16 + row];
        s2 = wmma4(a, b, s2);
      }
      v8f o;
#pragma unroll
      for (int j = 0; j < 8; ++j) o[j] = evM[j][d] * r[j] + esM[j] * s2[j];
      vTmp[d] = o;
    }
  }

  if constexpr (GATED) {
    v8f g, sn;
#pragma unroll
    for (int j = 0; j < 8; ++j) {
      g[j]  = sigmoid_f(sOut[0][j]);              // gates = first 16 scalars
      sn[j] = sOut[1][j] * sigmoid_f(sOut[1][j]); // silu on remaining scalars
    }
    outS = sn;
#pragma unroll
    for (int d = 0; d < 3; ++d) {
      v8f o;
#pragma unroll
      for (int j = 0; j < 8; ++j) o[j] = vTmp[d][j] * g[j];
      outV[d] = o;
    }
  } else {
    outS = sOut[0];
#pragma unroll
    for (int d = 0; d < 3; ++d) outV[d] = vTmp[d];
  }

  if constexpr (WRITEBACK) {
    // transpose D layout (M per-VGPR, N per-lane) back to row-major scratch
#pragma unroll
    for (int j = 0; j < 8; ++j) {
      const int M = mbase + j;
      s_in[M * SPAD + row] = outS[j];
      v_in[(0 * 16 + M) * SPAD + row] = outV[0][j];
      v_in[(1 * 16 + M) * SPAD + row] = outV[1][j];
      v_in[(2 * 16 + M) * SPAD + row] = outV[2][j];
    }
    __syncthreads();
  }
}

// ---------------------------------------------------------------------------
// Node embedding: (2x0e + 1x1o) (x) node_attr -> 16x0e + 16x1o
// ---------------------------------------------------------------------------
__global__ void embed_kernel(const float* __restrict__ node_s,
                             const float* __restrict__ node_v,
                             const float* __restrict__ nas,
                             const float* __restrict__ nav,
                             const float* __restrict__ Ws,
                             const float* __restrict__ bsv,
                             const float* __restrict__ Wv,
                             float* __restrict__ hs, float* __restrict__ hv,
                             int N) {
  const int n = blockIdx.x * blockDim.x + threadIdx.x;
  if (n >= N) return;
  const float s0 = node_s[n * 2 + 0], s1 = node_s[n * 2 + 1];
  const float v0 = node_v[n * 3 + 0], v1 = node_v[n * 3 + 1], v2 = node_v[n * 3 + 2];
  const float as = nas[n];
  const float a0 = nav[n * 3 + 0], a1 = nav[n * 3 + 1], a2 = nav[n * 3 + 2];
  const float ts0 = s0 * as, ts1 = s1 * as, ts2 = v0 * a0 + v1 * a1 + v2 * a2;
  const float av[3] = {a0, a1, a2};
  const float vv[3] = {v0, v1, v2};
#pragma unroll
  for (int o = 0; o < Hh; ++o) {
    hs[n * Hh + o] = ts0 * Ws[0 * Hh + o] + ts1 * Ws[1 * Hh + o] + ts2 * Ws[2 * Hh + o] + bsv[o];
#pragma unroll
    for (int d = 0; d < 3; ++d) {
      const float t0 = s0 * av[d], t1 = s1 * av[d], t2 = vv[d] * as;
      hv[(n * Hh + o) * 3 + d] = t0 * Wv[0 * Hh + o] + t1 * Wv[1 * Hh + o] + t2 * Wv[2 * Hh + o];
    }
  }
}

// ---------------------------------------------------------------------------
// Edge MLP + scatter-add aggregation. 4 waves/block, 16 edges/wave.
// ---------------------------------------------------------------------------
__global__ void __launch_bounds__(WAVES * 32)
edge_kernel(const float* __restrict__ hs, const float* __restrict__ hv,
            float* __restrict__ ag_s, float* __restrict__ ag_v,
            const float* __restrict__ eas, const float* __restrict__ eav,
            const int* __restrict__ senders, const int* __restrict__ receivers,
            const float* __restrict__ e1Ws, const float* __restrict__ e1bs,
            const float* __restrict__ e1Wv,
            const float* __restrict__ e2Ws, const float* __restrict__ e2bs,
            const float* __restrict__ e2Wv,
            const float* __restrict__ e3Ws, const float* __restrict__ e3bs,
            const float* __restrict__ e3Wv,
            int t, int E, float invDenom) {
  // weight arena offsets (floats)
  // W1S:0(2048) W1V:2048(1024) W2S:3072(1024) W2V:4096(512)
  // W3S:4608(512) W3V:5120(512) B1:5632(32) B2:5664(32) B3:5696(16)
  __shared__ float wsh[5712];
  __shared__ float sc_s[WAVES][16 * SPAD];
  __shared__ float sc_v[WAVES][3 * 16 * SPAD];
  __shared__ float sc_es[WAVES][16];
  __shared__ float sc_ev[WAVES][16 * 3];
  __shared__ int   sc_rcv[WAVES][16];

  const int tid = threadIdx.x, bdim = blockDim.x;
  copy_lds(wsh + 0,    e1Ws + (size_t)t * 2048, 2048, tid, bdim);
  copy_lds(wsh + 2048, e1Wv + (size_t)t * 1024, 1024, tid, bdim);
  copy_lds(wsh + 3072, e2Ws + (size_t)t * 1024, 1024, tid, bdim);
  copy_lds(wsh + 4096, e2Wv + (size_t)t * 512,  512,  tid, bdim);
  copy_lds(wsh + 4608, e3Ws + (size_t)t * 512,  512,  tid, bdim);
  copy_lds(wsh + 5120, e3Wv + (size_t)t * 512,  512,  tid, bdim);
  copy_lds(wsh + 5632, e1bs + (size_t)t * 32,   32,   tid, bdim);
  copy_lds(wsh + 5664, e2bs + (size_t)t * 32,   32,   tid, bdim);
  copy_lds(wsh + 5696, e3bs + (size_t)t * 16,   16,   tid, bdim);

  const int wave = tid >> 5;
  const int lane = tid & 31;
  const int e0   = blockIdx.x * (WAVES * 16) + wave * 16;
  const int side = lane >> 4;   // 0: sender features (cols 0..15), 1: receiver (16..31)
  const int le   = lane & 15;   // edge within tile
  const int e    = e0 + le;
  const int ec   = (e < E) ? e : (E - 1);  // clamp: keep EXEC full for WMMA
  const int nidx = side ? receivers[ec] : senders[ec];

  float* ss = sc_s[wave];
  float* vv = sc_v[wave];
  const float* hsp = hs + (size_t)nidx * Hh;
#pragma unroll
  for (int ch = 0; ch < Hh; ++ch) ss[le * SPAD + side * 16 + ch] = hsp[ch];
  const float* hvp = hv + (size_t)nidx * (Hh * 3);
#pragma unroll
  for (int ch = 0; ch < Hh; ++ch) {
#pragma unroll
    for (int d = 0; d < 3; ++d)
      vv[(d * 16 + le) * SPAD + side * 16 + ch] = hvp[ch * 3 + d];
  }
  if (side == 0) {
    sc_es[wave][le] = eas[ec];
    sc_ev[wave][le * 3 + 0] = eav[ec * 3 + 0];
    sc_ev[wave][le * 3 + 1] = eav[ec * 3 + 1];
    sc_ev[wave][le * 3 + 2] = eav[ec * 3 + 2];
    sc_rcv[wave][le] = receivers[ec];
  }
  __syncthreads();

  v8f sN; v8f vN[3];
  edge_layer<32, 32, true,  true >(wsh + 0,    wsh + 5632, wsh + 2048,
                                   ss, vv, sc_es[wave], sc_ev[wave], lane, sN, vN);
  edge_layer<16, 32, true,  true >(wsh + 3072, wsh + 5664, wsh + 4096,
                                   ss, vv, sc_es[wave], sc_ev[wave], lane, sN, vN);
  edge_layer<16, 16, false, false>(wsh + 4608, wsh + 5696, wsh + 5120,
                                   ss, vv, sc_es[wave], sc_ev[wave], lane, sN, vN);

  // segment-sum into receiver nodes (pre-scaled by 1/(E-1))
  const int mbase = (lane >> 4) * 8;
  const int col   = lane & 15;
#pragma unroll
  for (int j = 0; j < 8; ++j) {
    const int m  = mbase + j;
    const int ee = e0 + m;
    if (ee < E) {
      const int node = sc_rcv[wave][m];
      atomicAdd(&ag_s[(size_t)node * Hh + col], sN[j] * invDenom);
      float* agv = &ag_v[((size_t)node * Hh + col) * 3];
      atomicAdd(agv + 0, vN[0][j] * invDenom);
      atomicAdd(agv + 1, vN[1][j] * invDenom);
      atomicAdd(agv + 2, vN[2][j] * invDenom);
    }
  }
}

// ---------------------------------------------------------------------------
// Node update: tp(concat(agg, h), node_attr) -> lin -> residual add.
// ---------------------------------------------------------------------------
__global__ void node_kernel(const float* __restrict__ nas,
                            const float* __restrict__ nav,
                            const float* __restrict__ ndWs,
                            const float* __restrict__ ndbs,
                            const float* __restrict__ ndWv,
                            const float* __restrict__ ag_s,
                            const float* __restrict__ ag_v,
                            float* __restrict__ hs, float* __restrict__ hv,
                            int N, int t) {
  __shared__ float Ws[64 * 16];
  __shared__ float Wv[64 * 16];
  __shared__ float bsv[16];
  copy_lds(Ws,  ndWs + (size_t)t * 1024, 1024, threadIdx.x, blockDim.x);
  copy_lds(Wv,  ndWv + (size_t)t * 1024, 1024, threadIdx.x, blockDim.x);
  copy_lds(bsv, ndbs + (size_t)t * 16,   16,   threadIdx.x, blockDim.x);
  __syncthreads();

  const int n = blockIdx.x * blockDim.x + threadIdx.x;
  if (n >= N) return;

  float cs[32], cv0[32], cv1[32], cv2[32];
#pragma unroll
  for (int k = 0; k < Hh; ++k) {
    cs[k]      = ag_s[(size_t)n * Hh + k];
    cs[Hh + k] = hs[(size_t)n * Hh + k];
    cv0[k]      = ag_v[((size_t)n * Hh + k) * 3 + 0];
    cv1[k]      = ag_v[((size_t)n * Hh + k) * 3 + 1];
    cv2[k]      = ag_v[((size_t)n * Hh + k) * 3 + 2];
    cv0[Hh + k] = hv[((size_t)n * Hh + k) * 3 + 0];
    cv1[Hh + k] = hv[((size_t)n * Hh + k) * 3 + 1];
    cv2[Hh + k] = hv[((size_t)n * Hh + k) * 3 + 2];
  }
  const float as = nas[n];
  const float a0 = nav[n * 3 + 0], a1 = nav[n * 3 + 1], a2 = nav[n * 3 + 2];

  for (int o = 0; o < Hh; ++o) {
    float P = 0.f, Q0 = 0.f, Q1 = 0.f, Q2 = 0.f;
    float R = 0.f, S0 = 0.f, S1 = 0.f, S2 = 0.f;
#pragma unroll
    for (int k = 0; k < 32; ++k) {
      const float wst = Ws[k * 16 + o],        wsb = Ws[(32 + k) * 16 + o];
      const float wvt = Wv[k * 16 + o],        wvb = Wv[(32 + k) * 16 + o];
      P  += cs[k]  * wst;  R  += cs[k]  * wvt;
      Q0 += cv0[k] * wsb;  Q1 += cv1[k] * wsb;  Q2 += cv2[k] * wsb;
      S0 += cv0[k] * wvb;  S1 += cv1[k] * wvb;  S2 += cv2[k] * wvb;
    }
    hs[(size_t)n * Hh + o] += as * P + a0 * Q0 + a1 * Q1 + a2 * Q2 + bsv[o];
    hv[((size_t)n * Hh + o) * 3 + 0] += a0 * R + as * S0;
    hv[((size_t)n * Hh + o) * 3 + 1] += a1 * R + as * S1;
    hv[((size_t)n * Hh + o) * 3 + 2] += a2 * R + as * S2;
  }
}

__global__ void output_kernel(const float* __restrict__ hs,
                              const float* __restrict__ hv,
                              float* __restrict__ out, int N) {
  const int idx = blockIdx.x * blockDim.x + threadIdx.x;
  if (idx >= N * 64) return;
  const int n = idx >> 6;
  const int c = idx & 63;
  out[idx] = (c < Hh) ? hs[(size_t)n * Hh + c] : hv[(size_t)n * (Hh * 3) + (c - Hh)];
}

// ---------------------------------------------------------------------------
extern "C" void kernel_launch(void* const* d_in, const int* in_sizes, int n_in,
                              void* d_out, int out_size, void* d_ws, size_t ws_size,
                              hipStream_t stream) {
  const float* node_s      = (const float*)d_in[0];
  const float* node_v      = (const float*)d_in[1];
  const float* node_attr_s = (const float*)d_in[2];
  const float* node_attr_v = (const float*)d_in[3];
  const float* edge_attr_s = (const float*)d_in[4];
  const float* edge_attr_v = (const float*)d_in[5];
  const int*   senders     = (const int*)d_in[6];
  const int*   receivers   = (const int*)d_in[7];
  const float* emb_Ws = (const float*)d_in[8];
  const float* emb_bs = (const float*)d_in[9];
  const float* emb_Wv = (const float*)d_in[10];
  const float* e1_Ws = (const float*)d_in[11];
  const float* e1_bs = (const float*)d_in[12];
  const float* e1_Wv = (const float*)d_in[13];
  const float* e2_Ws = (const float*)d_in[14];
  const float* e2_bs = (const float*)d_in[15];
  const float* e2_Wv = (const float*)d_in[16];
  const float* e3_Ws = (const float*)d_in[17];
  const float* e3_bs = (const float*)d_in[18];
  const float* e3_Wv = (const float*)d_in[19];
  const float* nd_Ws = (const float*)d_in[20];
  const float* nd_bs = (const float*)d_in[21];
  const float* nd_Wv = (const float*)d_in[22];

  const int N = in_sizes[2];   // node_attr_s has N elements
  const int E = in_sizes[4];   // edge_attr_s has E elements

  float* ws   = (float*)d_ws;
  float* hs   = ws;                     // [N][16]
  float* hv   = ws + (size_t)N * 16;    // [N][16][3]
  float* ag_s = ws + (size_t)N * 64;    // [N][16]
  float* ag_v = ws + (size_t)N * 80;    // [N][16][3]

  embed_kernel<<<(N + 255) / 256, 256, 0, stream>>>(
      node_s, node_v, node_attr_s, node_attr_v, emb_Ws, emb_bs, emb_Wv,
      hs, hv, N);

  const float invDenom = 1.0f / (float)(E - 1);
  for (int t = 0; t < 3; ++t) {
    hipMemsetAsync(ag_s, 0, (size_t)N * 64 * sizeof(float), stream);
    edge_kernel<<<(E + WAVES * 16 - 1) / (WAVES * 16), WAVES * 32, 0, stream>>>(
        hs, hv, ag_s, ag_v, edge_attr_s, edge_attr_v, senders, receivers,
        e1_Ws, e1_bs, e1_Wv, e2_Ws, e2_bs, e2_Wv, e3_Ws, e3_bs, e3_Wv,
        t, E, invDenom);
    node_kernel<<<(N + 127) / 128, 128, 0, stream>>>(
        node_attr_s, node_attr_v, nd_Ws, nd_bs, nd_Wv, ag_s, ag_v, hs, hv, N, t);
  }

  output_kernel<<<(N * 64 + 255) / 256, 256, 0, stream>>>(hs, hv, (float*)d_out, N);
}